// Aggregator_23313082483396
// MI455X (gfx1250) — compile-verified
//
#include <hip/hip_runtime.h>
#include <hip/hip_bf16.h>
#include <stdint.h>

#ifndef __has_builtin
#define __has_builtin(x) 0
#endif

#define D_COLS 100
#define K_NEI  20
#define ROWS_PER_BLOCK 8      // 8 wave32 per 256-thread block, one output row per wave
#define ROW_BYTES 400         // 100 f32

// Async-LDS builtin takes (int4 AS1* global_src, int4 AS3* lds_dst, imm offset, imm cpol)
typedef int v4i __attribute__((ext_vector_type(4)));
typedef v4i __attribute__((address_space(1)))* gptr4_t;
typedef v4i __attribute__((address_space(3)))* lptr4_t;

#if __has_builtin(__builtin_amdgcn_global_load_async_to_lds_b128)
#define HAVE_ASYNC_LDS 1
#else
#define HAVE_ASYNC_LDS 0
#endif

template<int N>
__device__ __forceinline__ void wait_async_le() {
#if __has_builtin(__builtin_amdgcn_s_wait_asynccnt)
    __builtin_amdgcn_s_wait_asynccnt(N);
#else
    asm volatile("s_wait_asynccnt %0" :: "i"(N) : "memory");
#endif
    asm volatile("" ::: "memory");   // pin LDS reads below the wait
}

#if HAVE_ASYNC_LDS
__device__ __forceinline__ void async_copy16(const void* g, void* lds_generic) {
    // Launder through integers: strips const/generic; AS3 pointers are 32-bit
    // LDS byte offsets (low 32 bits of a generic __shared__ pointer, ISA 10.2).
    __builtin_amdgcn_global_load_async_to_lds_b128(
        (gptr4_t)(uintptr_t)g,
        (lptr4_t)(uint32_t)(uintptr_t)lds_generic,
        /*offset=*/0, /*cpol=*/0);
}

// Async loads complete in order: before consuming staged row k it suffices
// that ASYNCcnt <= K_NEI-1-k. Template recursion keeps the wait immediates ICE.
template<int K>
struct Acc {
    static __device__ __forceinline__ void run(const char* mybuf, int cbyte, float4& acc) {
        wait_async_le<K_NEI - 1 - K>();
        const float4 v = *(const float4*)(mybuf + K * ROW_BYTES + cbyte);
        acc.x += v.x; acc.y += v.y; acc.z += v.z; acc.w += v.w;
        Acc<K + 1>::run(mybuf, cbyte, acc);
    }
};
template<>
struct Acc<K_NEI> {
    static __device__ __forceinline__ void run(const char*, int, float4&) {}
};
#endif

// ---------------------------------------------------------------------------
// Kernel 1: node_emb rows. Wave handles one row; lanes 0..24 own float4 each.
// news row n   : sum_k entity_emb[news_entities[n,k]]  + all_embedding[n]
// entity row n : sum_k all_embedding[neigh_entities[n,k]] + all_embedding[n]
// (softmax over a singleton axis == 1.0, so the logit path is exactly dead)
// ---------------------------------------------------------------------------
__global__ __launch_bounds__(256) void node_kernel(
    const float* __restrict__ all_emb,
    const float* __restrict__ entity_emb,
    const int*   __restrict__ news_entities,
    const int*   __restrict__ neigh_entities,
    float*       __restrict__ node_out,
    int n_news, int n_entity)
{
#if HAVE_ASYNC_LDS
    __shared__ char ldsbuf[ROWS_PER_BLOCK * K_NEI * ROW_BYTES];   // 64 000 B
#endif
    const int wave = threadIdx.x >> 5;
    const int lane = threadIdx.x & 31;
    const int row  = blockIdx.x * ROWS_PER_BLOCK + wave;
    if (row >= n_news + n_entity) return;

    const bool  is_news = (row < n_news);
    const int   lrow    = is_news ? row : row - n_news;
    const int*  idxs    = (is_news ? news_entities : neigh_entities) + (size_t)lrow * K_NEI;
    const char* table   = (const char*)(is_news ? entity_emb : all_emb);
    const float* base   = all_emb + (size_t)lrow * D_COLS;

    const int  cbyte = (lane & 31) * 16;     // byte offset of this lane's float4
    const bool act   = (cbyte < ROW_BYTES);  // lanes 0..24 carry data

    float4 acc;
    if (act) acc = *(const float4*)((const char*)base + cbyte);
    else     acc = make_float4(0.f, 0.f, 0.f, 0.f);

#if HAVE_ASYNC_LDS
    char* mybuf = ldsbuf + (size_t)wave * K_NEI * ROW_BYTES;
    #pragma unroll
    for (int k = 0; k < K_NEI; ++k) {
        // gathered row index is wave-uniform -> SGPR, enables saddr addressing
        const int idx = __builtin_amdgcn_readfirstlane(idxs[k]);
        if (act)
            async_copy16(table + (size_t)idx * ROW_BYTES + cbyte,
                         mybuf + k * ROW_BYTES + cbyte);
    }
    if (act) Acc<0>::run(mybuf, cbyte, acc);
    else     wait_async_le<0>();    // keep per-wave ASYNCcnt drained uniformly
#else
    #pragma unroll
    for (int k = 0; k < K_NEI; ++k) {
        const int idx = __builtin_amdgcn_readfirstlane(idxs[k]);
        if (k + 1 < K_NEI) {
            const int idxn = idxs[k + 1];
            if (act) __builtin_prefetch(table + (size_t)idxn * ROW_BYTES + cbyte, 0, 3);
        }
        if (act) {
            const float4 v = *(const float4*)(table + (size_t)idx * ROW_BYTES + cbyte);
            acc.x += v.x; acc.y += v.y; acc.z += v.z; acc.w += v.w;
        }
    }
#endif

    if (act)
        *(float4*)((char*)(node_out + (size_t)row * D_COLS) + cbyte) = acc;
}

// ---------------------------------------------------------------------------
// Kernel 2: user_agg = user_emb (seed before atomic accumulation)
// ---------------------------------------------------------------------------
__global__ __launch_bounds__(256) void user_init_kernel(
    const float* __restrict__ user_emb, float* __restrict__ user_out, int n4)
{
    const int i = blockIdx.x * blockDim.x + threadIdx.x;
    if (i < n4) ((float4*)user_out)[i] = ((const float4*)user_emb)[i];
}

// ---------------------------------------------------------------------------
// Kernel 3: user_agg[rows[e]] += vals[e] * node_emb[cols[e]]
// one wave per nnz edge; lanes 0..24 float4-load + 4x global_atomic_add_f32
// ---------------------------------------------------------------------------
__global__ __launch_bounds__(256) void scatter_kernel(
    const float* __restrict__ vals,
    const int*   __restrict__ rows,
    const int*   __restrict__ cols,
    const float* __restrict__ node_emb,
    float*       __restrict__ user_out,
    int nnz)
{
    const int wave = threadIdx.x >> 5;
    const int lane = threadIdx.x & 31;
    const int e = blockIdx.x * ROWS_PER_BLOCK + wave;
    if (e >= nnz) return;
    const int cbyte = lane * 16;
    if (cbyte >= ROW_BYTES) return;

    // edge payload is wave-uniform: force into SGPRs for saddr addressing
    const int   r   = __builtin_amdgcn_readfirstlane(rows[e]);
    const int   col = __builtin_amdgcn_readfirstlane(cols[e]);
    const float v   = __int_as_float(__builtin_amdgcn_readfirstlane(__float_as_int(vals[e])));

    const float4 nv = *(const float4*)((const char*)(node_emb + (size_t)col * D_COLS) + cbyte);
    float* dst = (float*)((char*)(user_out + (size_t)r * D_COLS) + cbyte);
    atomicAdd(dst + 0, v * nv.x);
    atomicAdd(dst + 1, v * nv.y);
    atomicAdd(dst + 2, v * nv.z);
    atomicAdd(dst + 3, v * nv.w);
}

extern "C" void kernel_launch(void* const* d_in, const int* in_sizes, int n_in,
                              void* d_out, int out_size, void* d_ws, size_t ws_size,
                              hipStream_t stream)
{
    const float* user_emb       = (const float*)d_in[0];
    const float* all_emb        = (const float*)d_in[1];
    const float* entity_emb     = (const float*)d_in[2];
    // d_in[3..7] (relation_emb, W_news, b_news, W_ent, b_ent) are mathematically
    // irrelevant: softmax over a size-1 axis yields weight 1.0 exactly.
    const float* vals           = (const float*)d_in[8];
    const int*   news_entities  = (const int*)d_in[9];
    const int*   neigh_entities = (const int*)d_in[11];
    const int*   irows          = (const int*)d_in[13];
    const int*   icols          = (const int*)d_in[14];

    const int n_user   = in_sizes[0] / D_COLS;
    const int n_entity = in_sizes[2] / D_COLS;
    const int n_news   = in_sizes[9] / K_NEI;
    const int nnz      = in_sizes[8];
    const int n_nodes  = n_news + n_entity;

    float* node_out = (float*)d_out;
    float* user_out = (float*)d_out + (size_t)n_nodes * D_COLS;

    node_kernel<<<(n_nodes + ROWS_PER_BLOCK - 1) / ROWS_PER_BLOCK, 256, 0, stream>>>(
        all_emb, entity_emb, news_entities, neigh_entities, node_out, n_news, n_entity);

    const int n4 = n_user * D_COLS / 4;
    user_init_kernel<<<(n4 + 255) / 256, 256, 0, stream>>>(user_emb, user_out, n4);

    scatter_kernel<<<(nnz + ROWS_PER_BLOCK - 1) / ROWS_PER_BLOCK, 256, 0, stream>>>(
        vals, irows, icols, node_out, user_out, nnz);
}